// MACEInteraction_28879360098430
// MI455X (gfx1250) — compile-verified
//
#include <hip/hip_runtime.h>
#include <hip/hip_bf16.h>

typedef float v2f __attribute__((ext_vector_type(2)));
typedef float v8f __attribute__((ext_vector_type(8)));

#define C_ 128
#define N_ 16000
#define E_ 256000
#define A_ 10
#define RB_ 8
#define H_ 64
#define BE 64
#define INV_SQRT3 0.57735026918962584f
#define INV_SQRT_C 0.08838834764831845f      // 1/sqrt(128)
#define INV_SQRT_2C 0.0625f                  // 1/sqrt(256)
#define INV_SQRT_CA 0.027950849718747372f    // 1/sqrt(1280)

__device__ __forceinline__ float silu_f(float x) { return x / (1.0f + __expf(-x)); }

// D = A(16x4,f32) * B(4x16,f32) + C  -- full-precision WMMA on CDNA5
__device__ __forceinline__ v8f wmma_k4(v2f a, v2f b, v8f c) {
  return __builtin_amdgcn_wmma_f32_16x16x4_f32(false, a, false, b, (short)0, c,
                                               false, false);
}

__device__ __forceinline__ void atomAddF(float* p, float v) {
  unsafeAtomicAdd(p, v);  // global_atomic_add_f32 (A fits in L2 -> on-chip RMW)
}

// gfx1250 async global->LDS copy (ASYNCcnt-tracked, GV addressing)
__device__ __forceinline__ void async_g2lds_b32(const float* g, float* ldsp) {
  uint32_t l = (uint32_t)(uintptr_t)ldsp;  // LDS aperture: low 32 bits = LDS offset
  asm volatile("global_load_async_to_lds_b32 %0, %1, off"
               :: "v"(l), "v"(g) : "memory");
}
__device__ __forceinline__ void wait_async0() {
  asm volatile("s_wait_asynccnt 0" ::: "memory");
}

__global__ void zero_kernel(float4* __restrict__ p, int n4) {
  int i = blockIdx.x * blockDim.x + threadIdx.x;
  if (i < n4) p[i] = make_float4(0.f, 0.f, 0.f, 0.f);
}

// hcat[n, 0*128 + w] = (x0 @ Wpre0) * inv_sqrt_c   (contiguous-K A loads)
__global__ __launch_bounds__(256) void pre0_kernel(
    const float* __restrict__ nf, const float* __restrict__ W,
    float* __restrict__ hcat) {
  const int lane = threadIdx.x & 31, wave = threadIdx.x >> 5;
  const int laneM = lane & 15, half = lane >> 4;
  const int mBase = blockIdx.x * 16, nBase = wave * 16;
  const float* arow = nf + (size_t)(mBase + laneM) * 512;
  v8f acc = {};
#pragma unroll 8
  for (int k0 = 0; k0 < C_; k0 += 4) {
    const int ka = k0 + 2 * half;
    v2f a = *(const v2f*)(arow + ka);  // global_load_b64
    v2f b;
    b.x = W[ka * C_ + nBase + laneM];
    b.y = W[(ka + 1) * C_ + nBase + laneM];
    acc = wmma_k4(a, b, acc);
  }
#pragma unroll
  for (int r = 0; r < 8; ++r)
    hcat[(mBase + r + 8 * half) * 512 + nBase + laneM] = acc[r] * INV_SQRT_C;
}

// hcat[n, (d+1)*128 + w] = (x1[:,:,d] @ Wpre1) * inv_sqrt_c  (stride-3 A loads)
__global__ __launch_bounds__(256) void pre1_kernel(
    const float* __restrict__ nf, const float* __restrict__ W,
    float* __restrict__ hcat) {
  const int lane = threadIdx.x & 31, wave = threadIdx.x >> 5;
  const int laneM = lane & 15, half = lane >> 4;
  const int d = blockIdx.y;  // 0..2
  const int mBase = blockIdx.x * 16, nBase = wave * 16;
  const float* arow = nf + (size_t)(mBase + laneM) * 512 + 128 + d;
  v8f acc = {};
#pragma unroll 8
  for (int k0 = 0; k0 < C_; k0 += 4) {
    const int ka = k0 + 2 * half;
    v2f a, b;
    a.x = arow[3 * ka];
    a.y = arow[3 * ka + 3];
    b.x = W[ka * C_ + nBase + laneM];
    b.y = W[(ka + 1) * C_ + nBase + laneM];
    acc = wmma_k4(a, b, acc);
  }
#pragma unroll
  for (int r = 0; r < 8; ++r)
    hcat[(mBase + r + 8 * half) * 512 + (d + 1) * C_ + nBase + laneM] =
        acc[r] * INV_SQRT_C;
}

// Fused edge kernel: async-staged inputs, radial MLP (WMMA+SiLU via LDS),
// message formation, atomic segment-sum scatter. 64 edges / block, 8 waves.
__global__ __launch_bounds__(256) void edge_kernel(
    const float* __restrict__ sph, const float* __restrict__ rb,
    const int* __restrict__ eidx, const float* __restrict__ Wm1,
    const float* __restrict__ Wm2, const float* __restrict__ Wm3,
    const float* __restrict__ hcat, float* __restrict__ A0,
    float* __restrict__ A1) {
  __shared__ float rbS[BE * RB_];
  __shared__ float hm1S[BE * H_];
  __shared__ float hm2S[BE * H_];
  __shared__ float YS[BE * 4];
  __shared__ int sndS[BE], rcvS[BE];
  const int tid = threadIdx.x;
  const int lane = tid & 31, wave = tid >> 5;
  const int laneM = lane & 15, half = lane >> 4;
  const int eBlock = blockIdx.x * BE;
  // async global->LDS staging (tensor/async data path, no VGPR round-trip)
  async_g2lds_b32(rb + eBlock * RB_ + tid, &rbS[tid]);
  async_g2lds_b32(rb + eBlock * RB_ + 256 + tid, &rbS[256 + tid]);
  async_g2lds_b32(sph + eBlock * 4 + tid, &YS[tid]);
  if (tid < BE) {
    sndS[tid] = eidx[eBlock + tid];
    rcvS[tid] = eidx[E_ + eBlock + tid];
  }
  wait_async0();
  __syncthreads();
  // stage 1: hmlp1 = silu(rb @ Wm1)   (64x8)@(8x64)
#pragma unroll
  for (int t = wave; t < (BE / 16) * 4; t += 8) {
    const int mg = t >> 2, nt = t & 3;
    v8f acc = {};
#pragma unroll
    for (int k0 = 0; k0 < RB_; k0 += 4) {
      const int ka = k0 + 2 * half;
      v2f a = *(const v2f*)(&rbS[(mg * 16 + laneM) * RB_ + ka]);  // ds_load_b64
      v2f b;
      b.x = Wm1[ka * H_ + nt * 16 + laneM];
      b.y = Wm1[(ka + 1) * H_ + nt * 16 + laneM];
      acc = wmma_k4(a, b, acc);
    }
#pragma unroll
    for (int r = 0; r < 8; ++r)
      hm1S[(mg * 16 + r + 8 * half) * H_ + nt * 16 + laneM] = silu_f(acc[r]);
  }
  __syncthreads();
  // stage 2: hmlp2 = silu(hmlp1 @ Wm2)  (64x64)@(64x64)
  for (int t = wave; t < (BE / 16) * 4; t += 8) {
    const int mg = t >> 2, nt = t & 3;
    v8f acc = {};
#pragma unroll 8
    for (int k0 = 0; k0 < H_; k0 += 4) {
      const int ka = k0 + 2 * half;
      v2f a = *(const v2f*)(&hm1S[(mg * 16 + laneM) * H_ + ka]);
      v2f b;
      b.x = Wm2[ka * H_ + nt * 16 + laneM];
      b.y = Wm2[(ka + 1) * H_ + nt * 16 + laneM];
      acc = wmma_k4(a, b, acc);
    }
#pragma unroll
    for (int r = 0; r < 8; ++r)
      hm2S[(mg * 16 + r + 8 * half) * H_ + nt * 16 + laneM] = silu_f(acc[r]);
  }
  __syncthreads();
  // stage 3: w = hmlp2 @ Wm3 (4 z-slabs per channel tile) -> messages -> atomics
  for (int t = wave; t < (BE / 16) * 8; t += 8) {
    const int mg = t >> 3, ct = t & 7;
    v8f acc0 = {}, acc1 = {}, acc2 = {}, acc3 = {};
#pragma unroll 4
    for (int k0 = 0; k0 < H_; k0 += 4) {
      const int ka = k0 + 2 * half;
      v2f a = *(const v2f*)(&hm2S[(mg * 16 + laneM) * H_ + ka]);
      const float* B0 = Wm3 + ka * 512 + ct * 16 + laneM;
      const float* B1 = Wm3 + (ka + 1) * 512 + ct * 16 + laneM;
      v2f b;
      b.x = B0[0];   b.y = B1[0];   acc0 = wmma_k4(a, b, acc0);
      b.x = B0[128]; b.y = B1[128]; acc1 = wmma_k4(a, b, acc1);
      b.x = B0[256]; b.y = B1[256]; acc2 = wmma_k4(a, b, acc2);
      b.x = B0[384]; b.y = B1[384]; acc3 = wmma_k4(a, b, acc3);
    }
    const int c = ct * 16 + laneM;
#pragma unroll
    for (int r = 0; r < 8; ++r) {
      const int el = mg * 16 + r + 8 * half;
      const int snd = sndS[el], rcv = rcvS[el];
      const float Y0 = YS[el * 4 + 0], Y1 = YS[el * 4 + 1];
      const float Y2 = YS[el * 4 + 2], Y3 = YS[el * 4 + 3];
      const float* hrow = hcat + (size_t)snd * 512 + c;  // L2-resident gather
      const float e0  = hrow[0];
      const float e1x = hrow[128];
      const float e1y = hrow[256];
      const float e1z = hrow[384];
      const float w0 = acc0[r], w1 = acc1[r], w2 = acc2[r], w3 = acc3[r];
      float* a0p = A0 + (size_t)rcv * 256;
      float* a1p = A1 + (size_t)rcv * 768;
      atomAddF(a0p + c, w0 * e0 * Y0);
      atomAddF(a0p + 128 + c,
               w1 * (e1x * Y1 + e1y * Y2 + e1z * Y3) * INV_SQRT3);
      atomAddF(a1p + c * 3 + 0, w2 * e0 * Y1);
      atomAddF(a1p + c * 3 + 1, w2 * e0 * Y2);
      atomAddF(a1p + c * 3 + 2, w2 * e0 * Y3);
      atomAddF(a1p + (128 + c) * 3 + 0, w3 * e1x * Y0);
      atomAddF(a1p + (128 + c) * 3 + 1, w3 * e1y * Y0);
      atomAddF(a1p + (128 + c) * 3 + 2, w3 * e1z * Y0);
    }
  }
}

// ocat[n, 0*128 + w] = (A0 @ Wpost0) * inv_sqrt_2c   (contiguous-K A loads)
__global__ __launch_bounds__(256) void post0_kernel(
    const float* __restrict__ A0, const float* __restrict__ W,
    float* __restrict__ ocat) {
  const int lane = threadIdx.x & 31, wave = threadIdx.x >> 5;
  const int laneM = lane & 15, half = lane >> 4;
  const int mBase = blockIdx.x * 16, nBase = wave * 16;
  const float* arow = A0 + (size_t)(mBase + laneM) * 256;
  v8f acc = {};
#pragma unroll 8
  for (int k0 = 0; k0 < 2 * C_; k0 += 4) {
    const int ka = k0 + 2 * half;
    v2f a = *(const v2f*)(arow + ka);  // global_load_b64
    v2f b;
    b.x = W[ka * C_ + nBase + laneM];
    b.y = W[(ka + 1) * C_ + nBase + laneM];
    acc = wmma_k4(a, b, acc);
  }
#pragma unroll
  for (int r = 0; r < 8; ++r)
    ocat[(mBase + r + 8 * half) * 512 + nBase + laneM] = acc[r] * INV_SQRT_2C;
}

// ocat[n, (d+1)*128 + w] = (A1[:,:,d] @ Wpost1) * inv_sqrt_2c  (stride-3 A loads)
__global__ __launch_bounds__(256) void post1_kernel(
    const float* __restrict__ A1, const float* __restrict__ W,
    float* __restrict__ ocat) {
  const int lane = threadIdx.x & 31, wave = threadIdx.x >> 5;
  const int laneM = lane & 15, half = lane >> 4;
  const int d = blockIdx.y;  // 0..2
  const int mBase = blockIdx.x * 16, nBase = wave * 16;
  const float* arow = A1 + (size_t)(mBase + laneM) * 768 + d;
  v8f acc = {};
#pragma unroll 8
  for (int k0 = 0; k0 < 2 * C_; k0 += 4) {
    const int ka = k0 + 2 * half;
    v2f a, b;
    a.x = arow[3 * ka];
    a.y = arow[3 * ka + 3];
    b.x = W[ka * C_ + nBase + laneM];
    b.y = W[(ka + 1) * C_ + nBase + laneM];
    acc = wmma_k4(a, b, acc);
  }
#pragma unroll
  for (int r = 0; r < 8; ++r)
    ocat[(mBase + r + 8 * half) * 512 + (d + 1) * C_ + nBase + laneM] =
        acc[r] * INV_SQRT_2C;
}

// out[n, w, z] = einsum over K=C*A=1280 of (o_z ⊗ attr) @ Wa(z).reshape(1280,128)
__global__ __launch_bounds__(256) void attr_kernel(
    const float* __restrict__ ocat, const float* __restrict__ attr,
    const float* __restrict__ Wa0, const float* __restrict__ Wa1,
    float* __restrict__ out) {
  __shared__ float oS[16 * 128];
  __shared__ float attrS[16 * A_];
  const int tid = threadIdx.x;
  const int lane = tid & 31, wave = tid >> 5;
  const int laneM = lane & 15, half = lane >> 4;
  const int z = blockIdx.y;
  const int mBase = blockIdx.x * 16, nBase = wave * 16;
  for (int i = tid; i < 16 * 128; i += 256)
    oS[i] = ocat[(size_t)(mBase + (i >> 7)) * 512 + z * C_ + (i & 127)];
  for (int i = tid; i < 16 * A_; i += 256) attrS[i] = attr[mBase * A_ + i];
  __syncthreads();
  const float* W = (z == 0) ? Wa0 : Wa1;
  const float* orow = oS + laneM * 128;
  const float* vrow = attrS + laneM * A_;
  v8f acc = {};
#pragma unroll 8
  for (int k0 = 0; k0 < C_ * A_; k0 += 4) {
    const int ka = k0 + 2 * half;
    const int u0 = ka / A_, v0 = ka - u0 * A_;
    const int u1 = (ka + 1) / A_, v1 = (ka + 1) - u1 * A_;
    v2f a, b;
    a.x = orow[u0] * vrow[v0];
    a.y = orow[u1] * vrow[v1];
    b.x = W[ka * C_ + nBase + laneM];
    b.y = W[(ka + 1) * C_ + nBase + laneM];
    acc = wmma_k4(a, b, acc);
  }
#pragma unroll
  for (int r = 0; r < 8; ++r)
    out[(size_t)(mBase + r + 8 * half) * 512 + (nBase + laneM) * 4 + z] =
        acc[r] * INV_SQRT_CA;
}

extern "C" void kernel_launch(void* const* d_in, const int* in_sizes, int n_in,
                              void* d_out, int out_size, void* d_ws,
                              size_t ws_size, hipStream_t stream) {
  const float* nf    = (const float*)d_in[0];
  const float* attr  = (const float*)d_in[1];
  const float* sph   = (const float*)d_in[2];
  const float* rb    = (const float*)d_in[3];
  const int*   eidx  = (const int*)d_in[4];
  const float* Wpre0 = (const float*)d_in[5];
  const float* Wpre1 = (const float*)d_in[6];
  const float* Wm1   = (const float*)d_in[7];
  const float* Wm2   = (const float*)d_in[8];
  const float* Wm3   = (const float*)d_in[9];
  const float* Wpost0 = (const float*)d_in[10];
  const float* Wpost1 = (const float*)d_in[11];
  const float* Wa0   = (const float*)d_in[12];
  const float* Wa1   = (const float*)d_in[13];
  float* outp = (float*)d_out;

  float* hcat = (float*)d_ws;                      // N x 512
  float* A0   = hcat + (size_t)N_ * 512;           // N x 256
  float* A1   = A0 + (size_t)N_ * 256;             // N x 256 x 3
  float* ocat = A1 + (size_t)N_ * 768;             // N x 512

  // zero the segment-sum accumulators (N*1024 floats = N*256 float4)
  zero_kernel<<<N_ * 256 / 256, 256, 0, stream>>>((float4*)A0, N_ * 256);
  pre0_kernel<<<N_ / 16, 256, 0, stream>>>(nf, Wpre0, hcat);
  pre1_kernel<<<dim3(N_ / 16, 3), 256, 0, stream>>>(nf, Wpre1, hcat);
  edge_kernel<<<E_ / BE, 256, 0, stream>>>(sph, rb, eidx, Wm1, Wm2, Wm3, hcat,
                                           A0, A1);
  post0_kernel<<<N_ / 16, 256, 0, stream>>>(A0, Wpost0, ocat);
  post1_kernel<<<dim3(N_ / 16, 3), 256, 0, stream>>>(A1, Wpost1, ocat);
  attr_kernel<<<dim3(N_ / 16, 4), 256, 0, stream>>>(ocat, attr, Wa0, Wa1,
                                                    outp);
}